// AFPM_73014444032728
// MI455X (gfx1250) — compile-verified
//
#include <hip/hip_runtime.h>
#include <hip/hip_bf16.h>
#include <math.h>

// Problem constants (match reference)
#define BB   8
#define CC   64
#define HH   512
#define WW   512
#define KK   8
#define HID  64
#define NH   64          // H/K
#define NW   64          // W/K
#define LL   4096        // NH*NW
#define K2   64          // K*K

typedef float v2f __attribute__((ext_vector_type(2)));
typedef float v4f __attribute__((ext_vector_type(4)));
typedef float v8f __attribute__((ext_vector_type(8)));

__device__ __forceinline__ float gelu_erf(float v) {
    // exact-erf GELU, matches jax.nn.gelu(approximate=False)
    return 0.5f * v * (1.0f + erff(v * 0.70710678118654752f));
}

// ---------------------------------------------------------------------------
// Kernel 0: per-patch-position dynamic kernels pk[L,64] and bias pb[L].
// Block = 256 threads = 4 patch positions x 64 (hidden/output) units.
// ---------------------------------------------------------------------------
__global__ __launch_bounds__(256) void afpm_pkpb(
    const float* __restrict__ w1k, const float* __restrict__ b1k,
    const float* __restrict__ w2k, const float* __restrict__ b2k,
    const float* __restrict__ w1b, const float* __restrict__ b1b,
    const float* __restrict__ w2b, const float* __restrict__ b2b,
    float* __restrict__ pk_out, float* __restrict__ pb_out)
{
    __shared__ float hk[4 * HID];
    __shared__ float hb[4 * HID];

    const int tid  = threadIdx.x;
    const int lsub = tid >> 6;          // 0..3
    const int j    = tid & 63;          // hidden unit / output column
    const int l    = blockIdx.x * 4 + lsub;

    const int iy = l >> 6, ix = l & 63;
    const float py = iy * 8.0f + 4.0f;
    const float px = ix * 8.0f + 4.0f;
    const float dy = py - 256.0f, dx = px - 256.0f;
    const float d  = sqrtf(dy * dy + dx * dx) * (1.0f / 362.03867196751236f); // /max_d

    hk[lsub * HID + j] = gelu_erf(d * w1k[j] + b1k[j]);
    hb[lsub * HID + j] = gelu_erf(d * w1b[j] + b1b[j]);
    __syncthreads();

    // pk[l][k] = sum_j hk[j] * w2k[j][k] + b2k[k]
    float acc = b2k[j];
    #pragma unroll 8
    for (int jj = 0; jj < HID; ++jj)
        acc += hk[lsub * HID + jj] * w2k[jj * K2 + j];
    pk_out[(size_t)l * K2 + j] = acc;

    if (j == 0) {
        float accb = b2b[0];
        #pragma unroll 8
        for (int jj = 0; jj < HID; ++jj)
            accb += hb[lsub * HID + jj] * w2b[jj];
        pb_out[l] = accb;
    }
}

// ---------------------------------------------------------------------------
// Kernel 1: fused unfold -> feats -> 1x1 conv (WMMA f32 16x16x4) -> mod -> fold.
// One workgroup = (batch b, row of 16 patches) = 8x128 image strip x 64 ch,
// staged once into LDS via GLOBAL_LOAD_ASYNC_TO_LDS_B128 (ASYNCcnt path).
//
// Dynamic LDS layout (floats):
//   S_patch [c=64][ky=8][x=128]  65536   (offset 0)      natural image layout
//   S_feats [c=64][p=16]          1024   (offset 65536)
//   S_feats2[o=64][p=16]          1024   (offset 66560)
//   S_pk    [p=16][k=64 pad->68]  1088   (offset 67584)  16B-aligned rows
//   S_pb    [p=16]                  16   (offset 68672)
// total 68688 floats = 274752 B  (<= 320 KB CDNA5 workgroup LDS)
// ---------------------------------------------------------------------------
__global__ __launch_bounds__(256) void afpm_main(
    const float* __restrict__ x,
    const float* __restrict__ conv_w, const float* __restrict__ conv_b,
    const float* __restrict__ pk_ws,  const float* __restrict__ pb_ws,
    float* __restrict__ out)
{
    extern __shared__ float smem[];
    float* S_patch  = smem;
    float* S_feats  = smem + 65536;
    float* S_feats2 = smem + 66560;
    float* S_pk     = smem + 67584;   // row stride 68 floats (bank-rotating, 16B aligned)
    float* S_pb     = smem + 68672;

    const int tid = threadIdx.x;
    const int wg  = blockIdx.x;           // grid = B * NH * (NW/16) = 2048
    const int b   = wg >> 8;
    const int iy  = (wg >> 2) & 63;
    const int tw  = wg & 3;
    const int l0  = iy * NW + tw * 16;    // first patch index of this tile
    const int y0  = iy * KK;
    const int x0  = tw * 128;
    const size_t img_base = (size_t)b * CC * HH * WW;

    // ---- Phase L: async-stage 8x128x64ch strip into LDS (no VGPR round trip)
    for (int i = 0; i < 64; ++i) {
        int t  = i * 256 + tid;               // 16384 b128 tasks
        int xq = t & 31;                      // 16B column within 128-wide strip
        int ky = (t >> 5) & 7;
        int c  = t >> 8;
        const float* gp = x + img_base + ((size_t)c * HH + (y0 + ky)) * WW + x0 + xq * 4;
        unsigned long long ga = (unsigned long long)(size_t)gp;
        // LDS byte offset = low 32 bits of the generic LDS address (ISA 10.2)
        unsigned int lo = (unsigned int)(size_t)(const void*)(S_patch + ((c * 8 + ky) * 128 + xq * 4));
        asm volatile("global_load_async_to_lds_b128 %0, %1, off th:TH_LOAD_NT"
                     :: "v"(lo), "v"(ga) : "memory");
    }

    // ---- preload pk/pb tile (overlaps with async staging) ------------------
    #pragma unroll
    for (int r = 0; r < 4; ++r) {
        int t = r * 256 + tid;
        int p = t >> 6, k = t & 63;
        S_pk[p * 68 + k] = pk_ws[(size_t)(l0 + p) * K2 + k];
    }
    if (tid < 16) S_pb[tid] = pb_ws[l0 + tid];

    asm volatile("s_wait_asynccnt 0" ::: "memory");
    __syncthreads();

    // ---- Phase F: feats[c][p] = sum_k patch[c,p,k]*pk[p,k] + pb[p] --------
    {
        const int c  = tid >> 2;
        const int pg = tid & 3;
        #pragma unroll
        for (int pi = 0; pi < 4; ++pi) {
            int p = pg + pi * 4;
            const v4f* pkv = (const v4f*)(S_pk + p * 68);
            float acc = 0.0f;
            #pragma unroll
            for (int ky = 0; ky < 8; ++ky) {
                v4f a0 = *(const v4f*)(S_patch + (c * 8 + ky) * 128 + p * 8);
                v4f a1 = *(const v4f*)(S_patch + (c * 8 + ky) * 128 + p * 8 + 4);
                v4f w0 = pkv[ky * 2 + 0];
                v4f w1 = pkv[ky * 2 + 1];
                acc += a0.x * w0.x + a0.y * w0.y + a0.z * w0.z + a0.w * w0.w
                     + a1.x * w1.x + a1.y * w1.y + a1.z * w1.z + a1.w * w1.w;
            }
            S_feats[c * 16 + p] = acc + S_pb[p];
        }
    }
    __syncthreads();

    // ---- Phase W: feats2 = conv_w @ feats + conv_b via V_WMMA_F32_16X16X4_F32
    // 4 waves, wave w owns output rows o = w*16 .. w*16+15; N = 16 patches;
    // K = 64 channels consumed 4 at a time (16 WMMA ops per wave).
    if (tid < 128) {
        const int w    = tid >> 5;
        const int lane = tid & 31;
        const int half = lane >> 4;       // 0: lanes 0-15, 1: lanes 16-31
        const int mn   = lane & 15;       // A row (M) / B column (N)
        const float* Arow = conv_w + (size_t)(w * 16 + mn) * CC;

        v8f acc = {};
        #pragma unroll
        for (int kk = 0; kk < 16; ++kk) {
            const int cb = kk * 4;
            // ISA 16x4 f32 A layout: VGPR0 = K=0 (lanes 0-15) / K=2 (16-31),
            //                        VGPR1 = K=1 / K=3. B (4x16) mirrors it.
            const int k0 = cb + (half ? 2 : 0);
            const int k1 = cb + (half ? 3 : 1);
            v2f a, bt;
            a.x  = Arow[k0];
            a.y  = Arow[k1];
            bt.x = S_feats[k0 * 16 + mn];
            bt.y = S_feats[k1 * 16 + mn];
            acc = __builtin_amdgcn_wmma_f32_16x16x4_f32(
                      false, a, false, bt, (short)0, acc, false, false);
        }
        // C/D layout: VGPR i -> M = i (lanes 0-15) or i+8 (lanes 16-31), N = lane&15
        #pragma unroll
        for (int i = 0; i < 8; ++i) {
            int o = w * 16 + i + half * 8;
            S_feats2[o * 16 + mn] = acc[i] + conv_b[o];
        }
    }
    __syncthreads();

    // ---- Phase M: out = patch * feats2, folded back, NT b128 stores -------
    for (int i = 0; i < 64; ++i) {
        int t  = i * 256 + tid;
        int xq = t & 31;
        int ky = (t >> 5) & 7;
        int c  = t >> 8;
        int p  = xq >> 1;
        const float f = S_feats2[c * 16 + p];
        v4f v = *(const v4f*)(S_patch + (c * 8 + ky) * 128 + xq * 4);
        v.x *= f; v.y *= f; v.z *= f; v.w *= f;
        __builtin_nontemporal_store(
            v, (v4f*)(out + img_base + ((size_t)c * HH + (y0 + ky)) * WW + x0 + xq * 4));
    }
}

// ---------------------------------------------------------------------------
extern "C" void kernel_launch(void* const* d_in, const int* in_sizes, int n_in,
                              void* d_out, int out_size, void* d_ws, size_t ws_size,
                              hipStream_t stream)
{
    (void)in_sizes; (void)n_in; (void)out_size; (void)ws_size;

    const float* x      = (const float*)d_in[0];
    const float* w1k    = (const float*)d_in[1];
    const float* b1k    = (const float*)d_in[2];
    const float* w2k    = (const float*)d_in[3];
    const float* b2k    = (const float*)d_in[4];
    const float* w1b    = (const float*)d_in[5];
    const float* b1b    = (const float*)d_in[6];
    const float* w2b    = (const float*)d_in[7];
    const float* b2b    = (const float*)d_in[8];
    const float* conv_w = (const float*)d_in[9];
    const float* conv_b = (const float*)d_in[10];
    float* outp = (float*)d_out;

    float* pk_ws = (float*)d_ws;                 // L*64 floats = 1 MB
    float* pb_ws = pk_ws + (size_t)LL * K2;      // L floats

    afpm_pkpb<<<LL / 4, 256, 0, stream>>>(w1k, b1k, w2k, b2k,
                                          w1b, b1b, w2b, b2b, pk_ws, pb_ws);

    const size_t shmem = 68688u * sizeof(float); // 274752 B (< 320 KB WGP LDS)
    hipFuncSetAttribute(reinterpret_cast<const void*>(afpm_main),
                        hipFuncAttributeMaxDynamicSharedMemorySize, (int)shmem);
    afpm_main<<<BB * NH * (NW / 16), 256, shmem, stream>>>(
        x, conv_w, conv_b, pk_ws, pb_ws, outp);
}